// GraphSage_2534030704731
// MI455X (gfx1250) — compile-verified
//
#include <hip/hip_runtime.h>
#include <hip/hip_bf16.h>

// ---------------------------------------------------------------------------
// GraphSAGE forward on MI455X (gfx1250, wave32).
//   N=50000 nodes, E=800000 edges, F=128 features, 3 SAGE layers + final lin.
// Strategy: L2-resident fp32 atomic aggregation + fp32 WMMA (16x16x4) GEMMs.
// ---------------------------------------------------------------------------

#define N_NODES 50000
#define N_EDGES 800000
#define FEAT    128
#define LDSTRIDE 132   // 16x128 tile padded: bank = (4*r + k) % 64, conflict-free

typedef __attribute__((ext_vector_type(2))) float v2f;
typedef __attribute__((ext_vector_type(8))) float v8f;

// ---------------------------------------------------------------- utilities
__global__ __launch_bounds__(256) void zero_kernel(float* __restrict__ p, int n) {
    int i = blockIdx.x * blockDim.x + threadIdx.x;
    if (i < n) p[i] = 0.0f;
}

__global__ __launch_bounds__(256) void count_kernel(const int* __restrict__ dst,
                                                    float* __restrict__ deg, int e_cnt) {
    int e = blockIdx.x * blockDim.x + threadIdx.x;
    if (e < e_cnt) unsafeAtomicAdd(deg + dst[e], 1.0f);
}

__global__ __launch_bounds__(256) void invert_deg_kernel(float* __restrict__ deg, int n) {
    int i = blockIdx.x * blockDim.x + threadIdx.x;
    if (i < n) {
        float c = deg[i];
        deg[i] = 1.0f / fmaxf(c, 1.0f);
    }
}

// One wave32 per edge: gather full 128-float source row, atomic-add into dst row.
__global__ __launch_bounds__(256) void scatter_kernel(const int* __restrict__ src,
                                                      const int* __restrict__ dst,
                                                      const float* __restrict__ h,
                                                      float* __restrict__ agg, int e_cnt) {
    int e = blockIdx.x * 8 + (threadIdx.x >> 5);
    if (e >= e_cnt) return;
    int lane = threadIdx.x & 31;
    int s = src[e];
    int d = dst[e];
    const float4 v = *(const float4*)(h + (size_t)s * FEAT + lane * 4);
    float* p = agg + (size_t)d * FEAT + lane * 4;
    unsafeAtomicAdd(p + 0, v.x);
    unsafeAtomicAdd(p + 1, v.y);
    unsafeAtomicAdd(p + 2, v.z);
    unsafeAtomicAdd(p + 3, v.w);
}

// agg[i][:] *= deg_inv[i]  (sum -> mean)
__global__ __launch_bounds__(256) void scale_kernel(float* __restrict__ agg,
                                                    const float* __restrict__ deg, int n_elts) {
    int i = blockIdx.x * blockDim.x + threadIdx.x;
    if (i < n_elts) agg[i] *= deg[i >> 7];   // FEAT == 128
}

// ------------------------------------------------------------- WMMA GEMM
// out[16 rows x 128 cols] = relu(mean @ Wl^T + bl + h @ Wr^T)
// Workgroup: 128 threads = 4 waves; each wave owns two 16x16 N-tiles.
// A fragments come from LDS (padded), B fragments from the 64KB weight
// matrices (WGP$/L2 resident). fp32 WMMA keeps full reference precision.
__global__ __launch_bounds__(128) void sage_gemm_kernel(const float* __restrict__ mean,
                                                        const float* __restrict__ hin,
                                                        const float* __restrict__ Wl,
                                                        const float* __restrict__ bl,
                                                        const float* __restrict__ Wr,
                                                        float* __restrict__ hout) {
    __shared__ float smM[16 * LDSTRIDE];
    __shared__ float smH[16 * LDSTRIDE];

    const int rowbase = blockIdx.x * 16;
    const int t = threadIdx.x;

    // Cooperative stage of the 16x128 mean-tile and h-tile into LDS.
#pragma unroll
    for (int p = 0; p < 4; ++p) {
        int idx = p * 512 + t * 4;       // 2048 floats per tile
        int r = idx >> 7;                // / 128
        int c = idx & 127;
        float4 vm = *(const float4*)(mean + (size_t)(rowbase + r) * FEAT + c);
        float4 vh = *(const float4*)(hin  + (size_t)(rowbase + r) * FEAT + c);
        *(float4*)(&smM[r * LDSTRIDE + c]) = vm;
        *(float4*)(&smH[r * LDSTRIDE + c]) = vh;
    }
    __syncthreads();

    const int wave   = t >> 5;
    const int lane   = t & 31;
    const int r      = lane & 15;          // A-matrix row / B,N column select
    const int khalf  = (lane >> 4) << 1;   // lanes 16-31 hold K+2,K+3
    const int rowoff = (lane >> 4) << 3;   // D: lanes 16-31 hold rows M+8..M+15

#pragma unroll
    for (int nt = 0; nt < 2; ++nt) {
        const int nbase = (wave * 2 + nt) * 16;
        const int ncol  = nbase + r;
        v8f acc = {};
#pragma unroll
        for (int k = 0; k < FEAT; k += 4) {
            // A: 16x4 fp32 fragment (2 VGPRs/lane), LDS (bank-conflict free)
            v2f am = *(const v2f*)(&smM[r * LDSTRIDE + k + khalf]);
            // B: 4x16 fragment of Wl^T -> Wl[ncol][k..]
            v2f bm = *(const v2f*)(Wl + (size_t)ncol * FEAT + k + khalf);
            acc = __builtin_amdgcn_wmma_f32_16x16x4_f32(false, am, false, bm,
                                                        (short)0, acc, false, false);
            v2f ah = *(const v2f*)(&smH[r * LDSTRIDE + k + khalf]);
            v2f bh = *(const v2f*)(Wr + (size_t)ncol * FEAT + k + khalf);
            acc = __builtin_amdgcn_wmma_f32_16x16x4_f32(false, ah, false, bh,
                                                        (short)0, acc, false, false);
        }
        const float bias = bl[ncol];
#pragma unroll
        for (int v = 0; v < 8; ++v) {
            float val = acc[v] + bias;
            val = val > 0.0f ? val : 0.0f;  // relu (applied to all 3 SAGE layers)
            hout[(size_t)(rowbase + v + rowoff) * FEAT + ncol] = val;
        }
    }
}

// Final projection: out[i] = h[i,:] . Wf + bf   (D_OUT == 1)
__global__ __launch_bounds__(256) void final_kernel(const float* __restrict__ h,
                                                    const float* __restrict__ Wf,
                                                    const float* __restrict__ bf,
                                                    float* __restrict__ out, int n) {
    int i = blockIdx.x * blockDim.x + threadIdx.x;
    if (i >= n) return;
    const float4* hr = (const float4*)(h + (size_t)i * FEAT);
    const float4* wr = (const float4*)Wf;
    float s = 0.0f;
#pragma unroll
    for (int k = 0; k < FEAT / 4; ++k) {
        float4 a = hr[k];
        float4 b = wr[k];
        s += a.x * b.x + a.y * b.y + a.z * b.z + a.w * b.w;
    }
    out[i] = s + bf[0];
}

// ---------------------------------------------------------------------------
extern "C" void kernel_launch(void* const* d_in, const int* in_sizes, int n_in,
                              void* d_out, int out_size, void* d_ws, size_t ws_size,
                              hipStream_t stream) {
    (void)in_sizes; (void)n_in; (void)out_size; (void)ws_size;

    const float* x   = (const float*)d_in[0];
    const int*   ei  = (const int*)d_in[1];      // [2, E] flat: row0=src, row1=dst
    const float* Wl[3] = { (const float*)d_in[2], (const float*)d_in[5], (const float*)d_in[8] };
    const float* bl[3] = { (const float*)d_in[3], (const float*)d_in[6], (const float*)d_in[9] };
    const float* Wr[3] = { (const float*)d_in[4], (const float*)d_in[7], (const float*)d_in[10] };
    const float* Wf  = (const float*)d_in[11];
    const float* bf  = (const float*)d_in[12];
    float* out = (float*)d_out;

    const int* src = ei;
    const int* dst = ei + N_EDGES;

    // Workspace layout (floats): deg_inv[N] | agg[N*128] | hA[N*128] | hB[N*128]
    float* ws  = (float*)d_ws;
    float* deg = ws;
    float* agg = deg + N_NODES;
    float* hA  = agg + (size_t)N_NODES * FEAT;
    float* hB  = hA  + (size_t)N_NODES * FEAT;

    const int nfeat_elts = N_NODES * FEAT;

    // --- degree (shared across layers; same edge_index) ---
    zero_kernel<<<(N_NODES + 255) / 256, 256, 0, stream>>>(deg, N_NODES);
    count_kernel<<<(N_EDGES + 255) / 256, 256, 0, stream>>>(dst, deg, N_EDGES);
    invert_deg_kernel<<<(N_NODES + 255) / 256, 256, 0, stream>>>(deg, N_NODES);

    const float* hin = x;
    float* houts[3] = { hA, hB, hA };
    for (int l = 0; l < 3; ++l) {
        zero_kernel<<<(nfeat_elts + 255) / 256, 256, 0, stream>>>(agg, nfeat_elts);
        scatter_kernel<<<N_EDGES / 8, 256, 0, stream>>>(src, dst, hin, agg, N_EDGES);
        scale_kernel<<<(nfeat_elts + 255) / 256, 256, 0, stream>>>(agg, deg, nfeat_elts);
        sage_gemm_kernel<<<N_NODES / 16, 128, 0, stream>>>(agg, hin, Wl[l], bl[l], Wr[l], houts[l]);
        hin = houts[l];
    }

    final_kernel<<<(N_NODES + 255) / 256, 256, 0, stream>>>(hin, Wf, bf, out, N_NODES);
}